// CurveIntegrator_74002286510440
// MI455X (gfx1250) — compile-verified
//
#include <hip/hip_runtime.h>
#include <hip/hip_bf16.h>

// CurveIntegrator on gfx1250 (MI455X), v2.
// HBM-bound streaming kernel (182 MB moved, ~0.5 GFLOP). Four pixels per
// thread via 128-bit NT global loads/stores to cut vmem+SALU issue 4x per
// pixel (issue rate is co-critical with the ~8us HBM floor). Math stays fp32
// VALU: per-pixel 10x10 Lagrange operators are register-resident and unique
// per pixel, so WMMA/TDM paths cannot help.

typedef float v4f __attribute__((ext_vector_type(4)));

constexpr int KN   = 10;           // number of keypoints
constexpr int KBS  = 8;
constexpr int KH   = 180;
constexpr int KW   = 240;
constexpr int KHW  = KH * KW;      // 43200 (divisible by 4)
constexpr int NQ   = KHW / 4;      // 10800 float4 groups per channel plane
constexpr int NCO  = KN + 1;       // 11 output coeff channels

__global__ __launch_bounds__(256)
void curve_integrator_74002286510440_kernel(
    const float* __restrict__ deriv,    // [BS][N][HW]
    const float* __restrict__ blurry,   // [BS][1][HW]
    const float* __restrict__ keyp,     // [BS][N][HW]
    float* __restrict__ out)            // coeffs [BS][11][HW] ++ cache [BS][N][N][HW]
{
    const int q = blockIdx.x * blockDim.x + threadIdx.x;   // float4 group in plane
    if (q >= NQ) return;
    const int b  = blockIdx.y;                             // batch index (no int div)
    const int hw = q * 4;                                  // 16B-aligned pixel offset

    float* __restrict__ out_coeffs = out;                         // 8*11*43200
    float* __restrict__ out_cache  = out + KBS * NCO * KHW;       // 8*10*10*43200

    // ---- read-once inputs: 128-bit NT loads (each line touched once)
    v4f kp[KN], dv[KN];
#pragma unroll
    for (int i = 0; i < KN; ++i) {
        kp[i] = __builtin_nontemporal_load((const v4f*)(keyp  + (b * KN + i) * KHW + hw));
        dv[i] = __builtin_nontemporal_load((const v4f*)(deriv + (b * KN + i) * KHW + hw));
    }
    const v4f bl = __builtin_nontemporal_load((const v4f*)(blurry + b * KHW + hw));

    const v4f vzero = {0.0f, 0.0f, 0.0f, 0.0f};
    const v4f vone  = {1.0f, 1.0f, 1.0f, 1.0f};

    v4f acc[KN];
#pragma unroll
    for (int o = 0; o < KN; ++o) acc[o] = vzero;

    // 1/(o+1) integration constants
    const float inv_o[KN] = {1.0f, 1.0f/2.0f, 1.0f/3.0f, 1.0f/4.0f, 1.0f/5.0f,
                             1.0f/6.0f, 1.0f/7.0f, 1.0f/8.0f, 1.0f/9.0f, 1.0f/10.0f};

#pragma unroll
    for (int i = 0; i < KN; ++i) {
        const v4f ki = kp[i];

        // coefficients of prod_{j!=i}(x - k_j); poly[k] is coeff of x^k
        v4f poly[KN];
        poly[0] = vone;
#pragma unroll
        for (int k = 1; k < KN; ++k) poly[k] = vzero;

        v4f denom = vone;
#pragma unroll
        for (int j = 0; j < KN; ++j) {
            if (j == i) continue;
            const v4f kj = kp[j];
            denom *= (ki - kj);
            // c[k] = c[k-1] - kj*c[k] (old values; walk high->low), c[0] = -kj*c[0]
#pragma unroll
            for (int k = KN - 1; k >= 1; --k) poly[k] = poly[k - 1] - kj * poly[k];
            poly[0] = -kj * poly[0];
        }

        const v4f invd = 1.0f / denom;   // IEEE divide per component

        // cache[b][o][i][hw]; streaming write-once -> 128-bit NT stores
        float* __restrict__ cbase = out_cache + (b * KN * KN + i) * KHW + hw;
#pragma unroll
        for (int o = 0; o < KN; ++o) {
            const v4f val = poly[o] * (invd * inv_o[o]);
            __builtin_nontemporal_store(val, (v4f*)(cbase + o * (KN * KHW)));
            acc[o] += val * dv[i];       // coeffs[o] += cache[o][i] * deriv[i]
        }
    }

    // integral = sum_{o odd} 2*coeffs[o]/(o+2)
    v4f integral = (2.0f * (1.0f / 3.0f)) * acc[1];
    integral += (2.0f * (1.0f / 5.0f))  * acc[3];
    integral += (2.0f * (1.0f / 7.0f))  * acc[5];
    integral += (2.0f * (1.0f / 9.0f))  * acc[7];
    integral += (2.0f * (1.0f / 11.0f)) * acc[9];
    const v4f baseline = (2.0f * bl - integral) * 0.5f;

    float* __restrict__ obase = out_coeffs + b * NCO * KHW + hw;
    __builtin_nontemporal_store(baseline, (v4f*)obase);
#pragma unroll
    for (int o = 0; o < KN; ++o)
        __builtin_nontemporal_store(acc[o], (v4f*)(obase + (o + 1) * KHW));
}

extern "C" void kernel_launch(void* const* d_in, const int* in_sizes, int n_in,
                              void* d_out, int out_size, void* d_ws, size_t ws_size,
                              hipStream_t stream) {
    // setup_inputs() order: derivative, blurry, keypoints (all float32)
    const float* deriv  = (const float*)d_in[0];
    const float* blurry = (const float*)d_in[1];
    const float* keyp   = (const float*)d_in[2];
    float* out = (float*)d_out;   // coeffs (8*11*HW) followed by cache (8*10*10*HW)

    constexpr int threads = 256;                          // 8 wave32 per block
    const int blocks_x = (NQ + threads - 1) / threads;    // 43
    hipLaunchKernelGGL(curve_integrator_74002286510440_kernel,
                       dim3(blocks_x, KBS), dim3(threads), 0, stream,
                       deriv, blurry, keyp, out);
}